// GlycanMS2CleavageNN_56049323213758
// MI455X (gfx1250) — compile-verified
//
#include <hip/hip_runtime.h>
#include <hip/hip_bf16.h>
#include <math.h>

typedef float v2f __attribute__((ext_vector_type(2)));
typedef float v8f __attribute__((ext_vector_type(8)));

#define D_FEAT 128
#define OUT_FEAT 128

// ---------------------------------------------------------------------------
// Kernel 1: feat = feature @ w_feat (WMMA f32 16x16x4), fused gate GEMV
//   block = 256 threads = 8 waves; each wave computes a 16x128 output slab.
//   w_feat staged through LDS in two 64-row halves (2 x 32KB).
// ---------------------------------------------------------------------------
__global__ void __launch_bounds__(256, 1)
gemm_score_kernel(const float* __restrict__ feature,
                  const float* __restrict__ w_feat,
                  const float* __restrict__ w_gate,
                  float* __restrict__ feat_out,
                  float* __restrict__ score_out,
                  int n)
{
    __shared__ float lds_w[64 * 128];   // 32 KB: half of w_feat (64 K-rows)

    const int tid   = threadIdx.x;
    const int wave  = tid >> 5;
    const int lane  = tid & 31;
    const int lrow  = lane & 15;        // M index within tile (A) / N index (B,D)
    const int khalf = lane >> 4;        // 0: K={0,1}, 1: K={2,3} within a k-step

    const int row0  = blockIdx.x * 128 + wave * 16;   // first row of this wave's slab
    int arow = row0 + lrow;
    int arow_c = arow < n ? arow : (n - 1);           // clamp loads; stores guarded
    const float* aptr = feature + (size_t)arow_c * D_FEAT;

    v8f  acc[8];
    #pragma unroll
    for (int t = 0; t < 8; ++t) acc[t] = (v8f)0.0f;

    float gpart = 0.0f;

    #pragma unroll
    for (int h = 0; h < 2; ++h) {
        __syncthreads();
        // cooperative load of 64 rows of w_feat into LDS (8192 floats / 256 thr)
        #pragma unroll
        for (int i = 0; i < 32; ++i)
            lds_w[i * 256 + tid] = w_feat[h * 64 * 128 + i * 256 + tid];
        __syncthreads();

        // A fragments for this half: 16 k-steps of 4, this lane holds 2 floats each
        v2f areg[16];
        #pragma unroll
        for (int ks = 0; ks < 16; ++ks) {
            const int k = h * 64 + ks * 4 + khalf * 2;
            v2f a;
            a.x = aptr[k + 0];
            a.y = aptr[k + 1];
            areg[ks] = a;
            // fused gate GEMV (w_gate is tiny, scalar-cache resident)
            gpart += a.x * w_gate[k] + a.y * w_gate[k + 1];
        }

        // 8 column tiles x 16 k-steps of V_WMMA_F32_16X16X4_F32
        #pragma unroll
        for (int t = 0; t < 8; ++t) {
            const int cn = t * 16;
            #pragma unroll
            for (int ks = 0; ks < 16; ++ks) {
                const int kb = ks * 4 + khalf * 2;   // K row within this half
                v2f b;
                b.x = lds_w[(kb + 0) * 128 + cn + lrow];
                b.y = lds_w[(kb + 1) * 128 + cn + lrow];
                acc[t] = __builtin_amdgcn_wmma_f32_16x16x4_f32(
                    false, areg[ks], false, b, (short)0, acc[t], false, false);
            }
        }
    }

    // finish gate score: lane l covers K={4k,4k+1}, lane l+16 covers K={4k+2,4k+3}
    gpart += __shfl_down(gpart, 16, 32);
    if (khalf == 0 && arow < n)
        score_out[arow] = gpart;

    // store D tiles: VGPR r -> row row0 + r + khalf*8, col cn + lrow
    #pragma unroll
    for (int t = 0; t < 8; ++t) {
        const int col = t * 16 + lrow;
        #pragma unroll
        for (int r = 0; r < 8; ++r) {
            const int rr = row0 + khalf * 8 + r;
            if (rr < n)
                feat_out[(size_t)rr * OUT_FEAT + col] = acc[t][r];
        }
    }
}

// ---------------------------------------------------------------------------
// Kernel 2: CSR row offsets from sorted dst via lower_bound (no atomics)
// ---------------------------------------------------------------------------
__global__ void __launch_bounds__(256)
rowptr_kernel(const int* __restrict__ dst, int* __restrict__ row_start,
              int n, int e_count)
{
    int i = blockIdx.x * blockDim.x + threadIdx.x;
    if (i > n) return;
    int lo = 0, hi = e_count;
    while (lo < hi) {
        int mid = (lo + hi) >> 1;
        if (dst[mid] < i) lo = mid + 1; else hi = mid;
    }
    row_start[i] = lo;
}

// ---------------------------------------------------------------------------
// Kernel 3: per-destination segment softmax + weighted gather-sum.
//   one wave32 per dst node; lane l owns channels [4l, 4l+4) as a float4.
//   Pass 1: lane-parallel max + shfl reduce. Pass 2: serial edges, float4
//   gathers of feat[src] (L2-resident), exp-weighted accumulate.
// ---------------------------------------------------------------------------
__global__ void __launch_bounds__(256)
aggregate_kernel(const int* __restrict__ src,
                 const float* __restrict__ score,
                 const float* __restrict__ feat,
                 const int* __restrict__ row_start,
                 float* __restrict__ out,
                 int n)
{
    const int wave = threadIdx.x >> 5;
    const int lane = threadIdx.x & 31;
    const int node = blockIdx.x * 8 + wave;
    if (node >= n) return;

    const int s = row_start[node];
    const int e = row_start[node + 1];

    // pass 1: segment max (numerically-stable softmax)
    float m = -INFINITY;
    for (int i = s + lane; i < e; i += 32)
        m = fmaxf(m, score[src[i]]);
    #pragma unroll
    for (int off = 16; off > 0; off >>= 1)
        m = fmaxf(m, __shfl_xor(m, off, 32));
    if (m == -INFINITY) m = 0.0f;   // empty segment, matches reference

    // pass 2: exp-weighted accumulation of projected features
    float denom = 0.0f;
    float4 acc = make_float4(0.f, 0.f, 0.f, 0.f);
    const float4* featv = (const float4*)feat;   // 32 float4 per row
    for (int i = s; i < e; ++i) {
        const int sn = src[i];
        const float ex = __expf(score[sn] - m);
        denom += ex;
        const float4 f = featv[(size_t)sn * 32 + lane];
        acc.x += ex * f.x; acc.y += ex * f.y;
        acc.z += ex * f.z; acc.w += ex * f.w;
    }

    const float inv = (e > s) ? (1.0f / denom) : 0.0f;
    float4 o = make_float4(acc.x * inv, acc.y * inv, acc.z * inv, acc.w * inv);
    ((float4*)out)[(size_t)node * 32 + lane] = o;
}

// ---------------------------------------------------------------------------
extern "C" void kernel_launch(void* const* d_in, const int* in_sizes, int n_in,
                              void* d_out, int out_size, void* d_ws, size_t ws_size,
                              hipStream_t stream)
{
    const float* feature = (const float*)d_in[0];
    const int*   src     = (const int*)d_in[1];
    const int*   dst     = (const int*)d_in[2];
    const float* w_gate  = (const float*)d_in[3];
    const float* w_feat  = (const float*)d_in[4];

    const int n       = in_sizes[0] / D_FEAT;   // N nodes
    const int e_count = in_sizes[1];            // E edges
    float* out = (float*)d_out;

    // workspace layout: [score: N f32][row_start: N+1 i32][pad][feat: N*128 f32]
    char* ws = (char*)d_ws;
    float* score     = (float*)ws;
    int*   row_start = (int*)(ws + (size_t)n * sizeof(float));
    size_t feat_off  = (((size_t)n * 4 + (size_t)(n + 1) * 4) + 255) & ~(size_t)255;
    float* feat      = (float*)(ws + feat_off);

    const int gemm_blocks = (n + 127) / 128;
    gemm_score_kernel<<<gemm_blocks, 256, 0, stream>>>(
        feature, w_feat, w_gate, feat, score, n);

    rowptr_kernel<<<(n + 1 + 255) / 256, 256, 0, stream>>>(
        dst, row_start, n, e_count);

    aggregate_kernel<<<(n + 7) / 8, 256, 0, stream>>>(
        src, score, feat, row_start, out, n);
}